// LEMA_35502199669097
// MI455X (gfx1250) — compile-verified
//
#include <hip/hip_runtime.h>
#include <math.h>

// ---------- problem constants (fixed by the reference) ----------
#define BB      2
#define NF      20000     // fusion voxels per batch
#define NS      30000     // source voxels per batch
#define CH      128
#define NKEY    500
#define NSAMP   16
#define NROWS   (BB*NF)   // 40000
#define NKROWS  (BB*NKEY) // 1000

// d_out layout (floats), in reference return order
#define OUT_OFF_OUT    0          // 1000*128
#define OUT_OFF_KVI    128000     // 1000*3
#define OUT_OFF_HEAT   131000     // 40000
#define OUT_OFF_SC10   171000     // 40000*10
#define OUT_OFF_KCLS   571000     // 1000

typedef float v2f __attribute__((ext_vector_type(2)));
typedef float v8f __attribute__((ext_vector_type(8)));

// ---------------------------------------------------------------
// One 16x16 f32 tile C = A[Mbase..Mbase+15, 0..K) * B[0..K, Nbase..Nbase+15]
// via v_wmma_f32_16x16x4_f32.  A row-major (lda), B row-major (ldb).
// Loads are UNCONDITIONAL from clamped (always in-bounds) addresses;
// out-of-range contributions are zeroed with branchless selects so the
// inner loop has no EXEC manipulation around the WMMAs.
// C layout: element j of v8f at lane L: row = Mbase + j + 8*(L>>4), col = Nbase + (L&15)
// ---------------------------------------------------------------
__device__ inline v8f gemm_tile16(const float* __restrict__ A, int lda,
                                  int Mbase, int Mlimit,
                                  const float* __restrict__ B, int ldb,
                                  int Nbase, int Nvalid, int K)
{
    int lane = threadIdx.x & 31;
    int half = lane >> 4;
    int l    = lane & 15;
    v8f c = {0.f,0.f,0.f,0.f,0.f,0.f,0.f,0.f};
    int  arow = Mbase + l;
    bool aok  = arow < Mlimit;
    const float* arp = A + (size_t)(aok ? arow : Mbase) * (size_t)lda;
    bool bok  = l < Nvalid;
    int  bcol = Nbase + (bok ? l : 0);
    for (int k0 = 0; k0 < K; k0 += 4) {
        int ka = k0 + 2*half;
        v2f av = *(const v2f*)(arp + ka);                   // global_load_b64
        float bx = B[(size_t)ka       * ldb + bcol];
        float by = B[(size_t)(ka + 1) * ldb + bcol];
        v2f a, b;
        a.x = aok ? av.x : 0.f;                             // v_cndmask, no branch
        a.y = aok ? av.y : 0.f;
        b.x = bok ? bx   : 0.f;
        b.y = bok ? by   : 0.f;
        c = __builtin_amdgcn_wmma_f32_16x16x4_f32(false, a, false, b,
                                                  (short)0, c, false, false);
    }
    return c;
}

// monotone descending sort key: min-u64 extraction == jax.lax.top_k order
__device__ inline unsigned long long desc_key(float v, unsigned idx)
{
    unsigned u = __float_as_uint(v);
    u = (u & 0x80000000u) ? ~u : (u | 0x80000000u);   // ascending-order map
    unsigned d = ~u;                                   // descending
    return ((unsigned long long)d << 32) | (unsigned long long)idx;
}

// ---------------------------------------------------------------
// K1: h1 = relu((fusion_feat @ heat_w1) * g1 + b1)   (40000x128)
// ---------------------------------------------------------------
__global__ void heat1_kernel(const float* __restrict__ X,
                             const float* __restrict__ W1,
                             const float* __restrict__ G1,
                             const float* __restrict__ B1,
                             float* __restrict__ H)
{
    int wid  = threadIdx.x >> 5;
    int tile = blockIdx.x * 8 + wid;           // 20000 tiles = 2500 x 8
    int mtile = tile >> 3, ntile = tile & 7;
    int lane = threadIdx.x & 31, half = lane >> 4, l = lane & 15;
    v8f c = gemm_tile16(X, CH, mtile*16, NROWS, W1, CH, ntile*16, 16, CH);
    int col = ntile*16 + l;
    float g = G1[col], b = B1[col];
    #pragma unroll
    for (int j = 0; j < 8; ++j) {
        int row = mtile*16 + j + 8*half;
        float v = fmaxf(c[j]*g + b, 0.f);
        H[(size_t)row*CH + col] = v;
    }
}

// ---------------------------------------------------------------
// K2: scores10 = h1 @ heat_w2 + b2 ; heat_scores/cls/topk-keys
// ---------------------------------------------------------------
__global__ void heat2_kernel(const float* __restrict__ H,
                             const float* __restrict__ W2,
                             const float* __restrict__ B2,
                             float* __restrict__ dout,
                             int* __restrict__ cls,
                             unsigned long long* __restrict__ keys)
{
    int wid  = threadIdx.x >> 5;
    int tile = blockIdx.x * 8 + wid;
    if (tile >= NROWS/16) return;              // wave-uniform
    int lane = threadIdx.x & 31, half = lane >> 4, l = lane & 15;
    v8f c = gemm_tile16(H, CH, tile*16, NROWS, W2, 10, 0, 10, CH);
    float b2c = (l < 10) ? B2[l] : 0.f;
    float* sc10 = dout + OUT_OFF_SC10;
    float* hsc  = dout + OUT_OFF_HEAT;
    #pragma unroll
    for (int j = 0; j < 8; ++j) {
        int row = tile*16 + j + 8*half;
        float v = c[j] + b2c;
        if (l < 10) sc10[(size_t)row*10 + l] = v;
        float rv = (l < 10) ? v : -INFINITY;
        int   ri = l;
        #pragma unroll
        for (int m = 8; m >= 1; m >>= 1) {     // stays within 16-lane group
            float ov = __shfl_xor(rv, m, 32);
            int   oi = __shfl_xor(ri, m, 32);
            if (ov > rv || (ov == rv && oi < ri)) { rv = ov; ri = oi; }
        }
        if (l == 0) {
            hsc[row] = rv;
            cls[row] = ri;
            int b = row / NF;
            unsigned ib = (unsigned)(row - b*NF);
            keys[row] = desc_key(rv, ib);
        }
    }
}

// ---------------------------------------------------------------
// K3: exact top-500 per batch (descending value, lowest-index ties)
// ---------------------------------------------------------------
__device__ inline void scan_min(const unsigned long long* kb, int tid,
                                const unsigned* removed,
                                unsigned long long& mn, int& mj)
{
    mn = ~0ULL; mj = -1;
    for (int j = 0; ; ++j) {
        int i = tid + (j << 8);
        if (i >= NF) break;
        if ((removed[j >> 5] >> (j & 31)) & 1u) continue;
        unsigned long long k = kb[i];
        if (k < mn) { mn = k; mj = j; }
    }
}

__global__ void topk_kernel(const unsigned long long* __restrict__ keys,
                            const int* __restrict__ cls,
                            const int* __restrict__ fusion_coords,
                            int* __restrict__ tki,
                            float* __restrict__ keyworld,
                            float* __restrict__ dout)
{
    __shared__ unsigned long long wmin[8];
    __shared__ unsigned long long gbc;
    int b   = blockIdx.x;
    int tid = threadIdx.x;
    const unsigned long long* kb = keys + (size_t)b * NF;
    unsigned removed[3] = {0u, 0u, 0u};
    unsigned long long mymin; int myj;
    scan_min(kb, tid, removed, mymin, myj);

    for (int r = 0; r < NKEY; ++r) {
        unsigned long long v = mymin;
        #pragma unroll
        for (int m = 16; m >= 1; m >>= 1) {
            unsigned long long o = __shfl_xor(v, m, 32);
            v = (o < v) ? o : v;
        }
        if ((tid & 31) == 0) wmin[tid >> 5] = v;
        __syncthreads();
        if (tid == 0) {
            unsigned long long g = wmin[0];
            #pragma unroll
            for (int i = 1; i < 8; ++i) g = (wmin[i] < g) ? wmin[i] : g;
            gbc = g;
            unsigned idx = (unsigned)g;
            int row = b*NKEY + r;
            tki[row] = (int)idx;
            int c0 = fusion_coords[((size_t)b*NF + idx)*2 + 0];
            int c1 = fusion_coords[((size_t)b*NF + idx)*2 + 1];
            float wx = ((float)c0 + 0.5f) * 8.0f * 0.075f + (-54.0f);
            float wy = ((float)c1 + 0.5f) * 8.0f * 0.075f + (-54.0f);
            keyworld[row*2 + 0] = wx;
            keyworld[row*2 + 1] = wy;
            float* kvi = dout + OUT_OFF_KVI;
            kvi[row*3 + 0] = (float)b;
            kvi[row*3 + 1] = (float)c0;
            kvi[row*3 + 2] = (float)c1;
            (dout + OUT_OFF_KCLS)[row] = (float)cls[(size_t)b*NF + idx];
        }
        __syncthreads();
        unsigned long long g = gbc;
        if (mymin == g && myj >= 0) {          // unique owner (keys unique)
            removed[myj >> 5] |= (1u << (myj & 31));
            scan_min(kb, tid, removed, mymin, myj);
        }
        __syncthreads();
    }
}

// ---------------------------------------------------------------
// K4: radius kNN: exact ordered top-16 by (d2, idx), then weighted gather
// one wave per (key, source); 2000 waves
// ---------------------------------------------------------------
__global__ void knn_kernel(const float* __restrict__ feat_a,
                           const float* __restrict__ feat_b,
                           const int* __restrict__ coords_a,
                           const int* __restrict__ coords_b,
                           const float* __restrict__ keyworld,
                           const float* __restrict__ knn_w,
                           const float* __restrict__ knn_b,
                           float* __restrict__ mf)
{
    int wid  = threadIdx.x >> 5;
    int lane = threadIdx.x & 31;
    int t    = blockIdx.x * 8 + wid;           // 0..1999
    int s    = t & 1;
    int kg   = t >> 1;                         // 0..999
    int b    = kg / NKEY;
    const float* feat   = s ? feat_b   : feat_a;
    const int*   coords = s ? coords_b : coords_a;
    float stride = s ? 8.0f : 4.0f;
    float radius = s ? 4.8f : 2.4f;
    float r2 = radius * radius;
    float kx = keyworld[kg*2 + 0];
    float ky = keyworld[kg*2 + 1];

    unsigned long long arr[16];
    #pragma unroll
    for (int q = 0; q < 16; ++q) arr[q] = ~0ULL;
    unsigned long long curmax = ~0ULL;

    const int* cb = coords + (size_t)b * NS * 2;
    for (int i = lane; i < NS; i += 32) {
        int c0 = cb[i*2 + 0], c1 = cb[i*2 + 1];
        float wx = __fadd_rn(__fmul_rn(__fmul_rn(__fadd_rn((float)c0, 0.5f), stride), 0.075f), -54.0f);
        float wy = __fadd_rn(__fmul_rn(__fmul_rn(__fadd_rn((float)c1, 0.5f), stride), 0.075f), -54.0f);
        float dx = __fsub_rn(kx, wx);
        float dy = __fsub_rn(ky, wy);
        float d2 = __fadd_rn(__fmul_rn(dx, dx), __fmul_rn(dy, dy));
        if (d2 <= r2) {
            unsigned long long key =
                ((unsigned long long)__float_as_uint(d2) << 32) | (unsigned)i;
            if (key < curmax) {
                int mp = 0; unsigned long long mv = arr[0];
                #pragma unroll
                for (int q = 1; q < 16; ++q) if (arr[q] > mv) { mv = arr[q]; mp = q; }
                #pragma unroll
                for (int q = 0; q < 16; ++q) if (q == mp) arr[q] = key;
                mv = arr[0];
                #pragma unroll
                for (int q = 1; q < 16; ++q) mv = (arr[q] > mv) ? arr[q] : mv;
                curmax = mv;
            }
        }
    }

    // 16 rounds of wave-wide ordered min extraction
    unsigned long long ordered[16];
    #pragma unroll
    for (int k = 0; k < 16; ++k) {
        unsigned long long lmin = arr[0]; int lp = 0;
        #pragma unroll
        for (int q = 1; q < 16; ++q) if (arr[q] < lmin) { lmin = arr[q]; lp = q; }
        unsigned long long g = lmin;
        #pragma unroll
        for (int m = 16; m >= 1; m >>= 1) {
            unsigned long long o = __shfl_xor(g, m, 32);
            g = (o < g) ? o : g;
        }
        unsigned long long ball = __ballot(lmin == g);
        int winner = (int)__builtin_ctzll(ball);
        if (lane == winner) {
            #pragma unroll
            for (int q = 0; q < 16; ++q) if (q == lp) arr[q] = ~0ULL;
        }
        ordered[k] = g;
    }

    // weighted gather: mf[c] = sum_k valid_k * knn_w[s][k] * feat[idx_k][c] + knn_b[s]
    float acc0 = 0.f, acc1 = 0.f, acc2 = 0.f, acc3 = 0.f;
    const float* fb = feat + (size_t)b * NS * CH;
    #pragma unroll
    for (int k = 0; k < 16; ++k) {
        unsigned long long g = ordered[k];
        bool valid = (unsigned)(g >> 32) < 0x7f800000u;   // finite d2
        if (valid) {                                       // wave-uniform
            float wk = knn_w[s*16 + k];
            const float* fr = fb + (size_t)((unsigned)g) * CH;
            acc0 += wk * fr[lane];
            acc1 += wk * fr[lane + 32];
            acc2 += wk * fr[lane + 64];
            acc3 += wk * fr[lane + 96];
        }
    }
    float bs = knn_b[s];
    float* mfr = mf + ((size_t)s * NKROWS + kg) * CH;
    mfr[lane]      = acc0 + bs;
    mfr[lane + 32] = acc1 + bs;
    mfr[lane + 64] = acc2 + bs;
    mfr[lane + 96] = acc3 + bs;
}

// ---------------------------------------------------------------
// K5: kw MLP (128->64->3) + softmax + cumulative combine -> fused(1000x256)
// one wave per key
// ---------------------------------------------------------------
__global__ void kw_kernel(const float* __restrict__ fusion_feat,
                          const int* __restrict__ tki,
                          const float* __restrict__ W1,
                          const float* __restrict__ G1,
                          const float* __restrict__ B1,
                          const float* __restrict__ W2,
                          const float* __restrict__ B2,
                          const float* __restrict__ mf,
                          float* __restrict__ fused)
{
    int wid  = threadIdx.x >> 5;
    int lane = threadIdx.x & 31;
    int kg   = blockIdx.x * 8 + wid;           // 0..999
    int b    = kg / NKEY;
    int row  = b*NF + tki[kg];
    const float* kf = fusion_feat + (size_t)row * CH;

    int na = lane, nb = lane + 32;
    float ha = 0.f, hb = 0.f;
    for (int k = 0; k < CH; ++k) {
        float x = kf[k];
        ha += x * W1[k*64 + na];
        hb += x * W1[k*64 + nb];
    }
    ha = fmaxf(ha * G1[na] + B1[na], 0.f);
    hb = fmaxf(hb * G1[nb] + B1[nb], 0.f);
    float p0 = ha * W2[na*3 + 0] + hb * W2[nb*3 + 0];
    float p1 = ha * W2[na*3 + 1] + hb * W2[nb*3 + 1];
    float p2 = ha * W2[na*3 + 2] + hb * W2[nb*3 + 2];
    #pragma unroll
    for (int m = 16; m >= 1; m >>= 1) {
        p0 += __shfl_xor(p0, m, 32);
        p1 += __shfl_xor(p1, m, 32);
        p2 += __shfl_xor(p2, m, 32);
    }
    p0 += B2[0]; p1 += B2[1]; p2 += B2[2];
    float mx = fmaxf(p0, fmaxf(p1, p2));
    float e0 = expf(p0 - mx), e1 = expf(p1 - mx), e2 = expf(p2 - mx);
    float inv = 1.f / (e0 + e1 + e2);
    float w0 = e0 * inv, w1 = e1 * inv;

    const float* mfa = mf + (size_t)kg * CH;
    const float* mfb = mf + (size_t)(NKROWS + kg) * CH;
    float* fr = fused + (size_t)kg * 256;
    #pragma unroll
    for (int q = 0; q < 4; ++q) {
        int c = lane + q*32;
        float x    = kf[c];
        float kvf1 = w0 * x;
        float fs   = kvf1 * mfa[c];
        float kvf2 = w1 * kvf1;
        fs += kvf2 * mfb[c];
        fr[c]       = fs;
        fr[128 + c] = kvf2;
    }
}

// ---------------------------------------------------------------
// K6: hf = relu((fused @ fuse_w1) * g1 + b1)   (1000x256)
// ---------------------------------------------------------------
__global__ void fuse1_kernel(const float* __restrict__ X,
                             const float* __restrict__ W1,
                             const float* __restrict__ G1,
                             const float* __restrict__ B1,
                             float* __restrict__ H)
{
    int wid  = threadIdx.x >> 5;
    int tile = blockIdx.x * 8 + wid;           // 63*16 = 1008 tiles
    int mtile = tile / 16, ntile = tile % 16;
    int lane = threadIdx.x & 31, half = lane >> 4, l = lane & 15;
    v8f c = gemm_tile16(X, 256, mtile*16, NKROWS, W1, 256, ntile*16, 16, 256);
    int col = ntile*16 + l;
    float g = G1[col], b = B1[col];
    #pragma unroll
    for (int j = 0; j < 8; ++j) {
        int row = mtile*16 + j + 8*half;
        if (row < NKROWS) H[(size_t)row*256 + col] = fmaxf(c[j]*g + b, 0.f);
    }
}

// ---------------------------------------------------------------
// K7: out = hf @ fuse_w2 + b2   (1000x128) -> d_out[0:128000]
// ---------------------------------------------------------------
__global__ void fuse2_kernel(const float* __restrict__ H,
                             const float* __restrict__ W2,
                             const float* __restrict__ B2,
                             float* __restrict__ dout)
{
    int wid  = threadIdx.x >> 5;
    int tile = blockIdx.x * 8 + wid;           // 63*8 = 504 tiles
    int mtile = tile >> 3, ntile = tile & 7;
    int lane = threadIdx.x & 31, half = lane >> 4, l = lane & 15;
    v8f c = gemm_tile16(H, 256, mtile*16, NKROWS, W2, CH, ntile*16, 16, 256);
    int col = ntile*16 + l;
    float b = B2[col];
    float* outp = dout + OUT_OFF_OUT;
    #pragma unroll
    for (int j = 0; j < 8; ++j) {
        int row = mtile*16 + j + 8*half;
        if (row < NKROWS) outp[(size_t)row*CH + col] = c[j] + b;
    }
}

// ---------------------------------------------------------------
extern "C" void kernel_launch(void* const* d_in, const int* in_sizes, int n_in,
                              void* d_out, int out_size, void* d_ws, size_t ws_size,
                              hipStream_t stream)
{
    (void)in_sizes; (void)n_in; (void)out_size; (void)ws_size;
    const float* fusion_feat = (const float*)d_in[0];
    const float* src_feat_a  = (const float*)d_in[1];
    const float* src_feat_b  = (const float*)d_in[2];
    const float* heat_w1 = (const float*)d_in[3];
    const float* heat_g1 = (const float*)d_in[4];
    const float* heat_b1 = (const float*)d_in[5];
    const float* heat_w2 = (const float*)d_in[6];
    const float* heat_b2 = (const float*)d_in[7];
    const float* knn_w   = (const float*)d_in[8];
    const float* knn_b   = (const float*)d_in[9];
    const float* kw_w1   = (const float*)d_in[10];
    const float* kw_g1   = (const float*)d_in[11];
    const float* kw_b1   = (const float*)d_in[12];
    const float* kw_w2   = (const float*)d_in[13];
    const float* kw_b2   = (const float*)d_in[14];
    const float* fuse_w1 = (const float*)d_in[15];
    const float* fuse_g1 = (const float*)d_in[16];
    const float* fuse_b1 = (const float*)d_in[17];
    const float* fuse_w2 = (const float*)d_in[18];
    const float* fuse_b2 = (const float*)d_in[19];
    const int* fusion_coords = (const int*)d_in[20];
    const int* src_coords_a  = (const int*)d_in[21];
    const int* src_coords_b  = (const int*)d_in[22];
    float* dout = (float*)d_out;

    // workspace carve (bytes)
    char* ws = (char*)d_ws;
    float*              h1       = (float*)(ws + 0);                         // 40000*128 f32
    unsigned long long* keys     = (unsigned long long*)(ws + 20480000);     // 40000 u64
    int*                cls      = (int*)(ws + 20800000);                    // 40000 i32
    int*                tki      = (int*)(ws + 20960000);                    // 1000 i32
    float*              keyworld = (float*)(ws + 20964000);                  // 1000*2 f32
    float*              mf       = (float*)(ws + 20972000);                  // 2*1000*128 f32
    float*              fused    = (float*)(ws + 21996000);                  // 1000*256 f32
    float*              hfuse    = (float*)(ws + 23020000);                  // 1000*256 f32

    heat1_kernel<<<2500, 256, 0, stream>>>(fusion_feat, heat_w1, heat_g1, heat_b1, h1);
    heat2_kernel<<<313, 256, 0, stream>>>(h1, heat_w2, heat_b2, dout, cls, keys);
    topk_kernel<<<BB, 256, 0, stream>>>(keys, cls, fusion_coords, tki, keyworld, dout);
    knn_kernel<<<250, 256, 0, stream>>>(src_feat_a, src_feat_b, src_coords_a, src_coords_b,
                                        keyworld, knn_w, knn_b, mf);
    kw_kernel<<<125, 256, 0, stream>>>(fusion_feat, tki, kw_w1, kw_g1, kw_b1,
                                       kw_w2, kw_b2, mf, fused);
    fuse1_kernel<<<126, 256, 0, stream>>>(fused, fuse_w1, fuse_g1, fuse_b1, hfuse);
    fuse2_kernel<<<63, 256, 0, stream>>>(hfuse, fuse_w2, fuse_b2, dout);
}